// DGM_d_33243046871166
// MI455X (gfx1250) — compile-verified
//
#include <hip/hip_runtime.h>

typedef __attribute__((ext_vector_type(2))) float v2f;
typedef __attribute__((ext_vector_type(8))) float v8f;

#define BB 4
#define NN 4096
#define DD 64
#define KSEL 10
#define NT (NN / 16)          // 256 column tiles
#define BSTRIDE 68            // padded LDS row stride for B tile (16B aligned, bank-spread)
#define DSTRIDE 20            // padded LDS row stride for lq transpose tile

// Async global->LDS copy (CDNA5): 16B per lane, tracked by ASYNCcnt.
__device__ __forceinline__ void async_copy_b128(void* lds_ptr, const void* gptr) {
  const unsigned l = (unsigned)(uintptr_t)lds_ptr;            // low 32 bits == LDS offset
  const unsigned long long g = (unsigned long long)(uintptr_t)gptr;
  asm volatile("global_load_async_to_lds_b128 %0, %1, off" ::"v"(l), "v"(g) : "memory");
}

// ---------------------------------------------------------------------------
// Kernel 1: x_emb = x @ W via V_WMMA_F32_16X16X4_F32 (full f32 precision).
// ---------------------------------------------------------------------------
__global__ __launch_bounds__(128) void dgm_embed_wmma(const float* __restrict__ x,
                                                      const float* __restrict__ W,
                                                      float* __restrict__ xemb) {
  const int lane = threadIdx.x & 31;
  const int wave = threadIdx.x >> 5;
  const int row0 = blockIdx.x * 16;
  const int col0 = wave * 16;
  const int m  = lane & 15;
  const int hi = lane >> 4;
  v8f acc = {};
#pragma unroll
  for (int s = 0; s < 16; ++s) {
    const int k = 4 * s + hi * 2;
    v2f a, b;
    const float* arow = x + (size_t)(row0 + m) * DD + k;
    a.x = arow[0];
    a.y = arow[1];
    b.x = W[(size_t)k * DD + col0 + m];
    b.y = W[(size_t)(k + 1) * DD + col0 + m];
    acc = __builtin_amdgcn_wmma_f32_16x16x4_f32(false, a, false, b, (short)0, acc,
                                                false, false);
  }
#pragma unroll
  for (int v = 0; v < 8; ++v) {
    const int r = v + 8 * hi;
    xemb[(size_t)(row0 + r) * DD + col0 + m] = acc[v];
  }
}

// ---------------------------------------------------------------------------
// Kernel 2: x2[row] = ||x_emb[row]||^2
// ---------------------------------------------------------------------------
__global__ void dgm_rownorm(const float* __restrict__ xemb, float* __restrict__ x2) {
  const int row = blockIdx.x * blockDim.x + threadIdx.x;
  if (row >= BB * NN) return;
  const float* p = xemb + (size_t)row * DD;
  float s = 0.f;
#pragma unroll
  for (int k = 0; k < DD; ++k) { const float v = p[k]; s += v * v; }
  x2[row] = s;
}

// ---------------------------------------------------------------------------
// Kernel 3 (fused): async-staged B tiles -> f32 WMMA Gram -> lq -> top-10.
// Block = 256 (8 waves) lock-stepped over j-tiles; B tile (4KB, contiguous)
// staged once per block via global_load_async_to_lds_b128, double buffered.
// q streamed exactly once with prefetch; lq never touches global memory.
// ---------------------------------------------------------------------------
__global__ __launch_bounds__(256) void dgm_topk(const float* __restrict__ xemb,
                                                const float* __restrict__ x2,
                                                const float* __restrict__ q,
                                                const float* __restrict__ temp,
                                                float* __restrict__ src_out,
                                                float* __restrict__ tgt_out,
                                                float* __restrict__ lp_out) {
  __shared__ float bbuf[2][16 * BSTRIDE];      // double-buffered B tile (async dest)
  __shared__ float dtile[8][16 * DSTRIDE];     // per-wave lq transpose tile

  const int tid  = threadIdx.x;
  const int lane = tid & 31;
  const int wave = tid >> 5;
  const int gw = blockIdx.x * 8 + wave;        // 0..1023
  const int i0 = gw * 16;                      // global row base (b*N + local)
  const int b  = i0 / NN;
  const int m  = lane & 15;
  const int hi = lane >> 4;
  float* tile = &dtile[wave][0];

  float tq = temp[0];
  tq = fminf(fmaxf(tq, -5.f), 5.f);
  const float texp = __expf(tq);

  // Preload A fragments (rows i0..i0+15) — resident in VGPRs for all 256 tiles
  v2f afr[16];
#pragma unroll
  for (int s = 0; s < 16; ++s) {
    const float* arow = xemb + (size_t)(i0 + m) * DD + 4 * s + hi * 2;
    afr[s].x = arow[0];
    afr[s].y = arow[1];
  }
  float x2i[8];
#pragma unroll
  for (int v = 0; v < 8; ++v) x2i[v] = x2[i0 + v + 8 * hi];

  float bestv[KSEL];
  int   bestj[KSEL];
#pragma unroll
  for (int t = 0; t < KSEL; ++t) { bestv[t] = 3.4e38f; bestj[t] = 0; }

  const float* xb = xemb + (size_t)b * NN * DD;
  const int cr = tid >> 4;           // staging: row 0..15
  const int cc = (tid & 15) * 4;     // staging: col 0,4,..,60

  // Prologue: kick off async stage of tile 0
  async_copy_b128(&bbuf[0][cr * BSTRIDE + cc], xb + (size_t)cr * DD + cc);

  for (int jt = 0; jt < NT; ++jt) {
    const int j0 = jt * 16;
    // Tile jt resident: my asyncs done, then block-wide rendezvous.
    asm volatile("s_wait_asynccnt 0x0" ::: "memory");
    __syncthreads();
    // Issue async stage of tile jt+1 into the other buffer (race-free: the
    // barrier above proves every wave finished reading it last iteration).
    if (jt + 1 < NT) {
      async_copy_b128(&bbuf[(jt + 1) & 1][cr * BSTRIDE + cc],
                      xb + (size_t)(j0 + 16 + cr) * DD + cc);
    }

    // q loads up front so they overlap the WMMA chain (one-pass HBM stream)
    float qv[8];
#pragma unroll
    for (int v = 0; v < 8; ++v)
      qv[v] = q[(size_t)(i0 + v + 8 * hi) * NN + j0 + m];
    const float x2j = x2[b * NN + j0 + m];
    if (((jt & 1) == 0) && (hi == 0) && (j0 + 32 < NN))
      __builtin_prefetch(&q[(size_t)(i0 + m) * NN + j0 + 32], 0, 1);

    // Gram tile from LDS-staged B: 16 chained f32 WMMAs
    const float* bbw = &bbuf[jt & 1][0];
    v8f acc = {};
#pragma unroll
    for (int s = 0; s < 16; ++s) {
      v2f bf;
      bf.x = bbw[m * BSTRIDE + 4 * s + 2 * hi];
      bf.y = bbw[m * BSTRIDE + 4 * s + 2 * hi + 1];
      acc = __builtin_amdgcn_wmma_f32_16x16x4_f32(false, afr[s], false, bf, (short)0,
                                                  acc, false, false);
    }

    // lq = (x2_i + x2_j - 2G) * e^T - ln(-ln(q+eps));  2x v_log_f32 per element
#pragma unroll
    for (int v = 0; v < 8; ++v) {
      const int M = v + 8 * hi;
      const float dist = x2i[v] + x2j - 2.f * acc[v];
      const float t1 = -__log2f(qv[v] + 1e-8f);
      const float gum = 0.6931471805599453f * __log2f(t1) - 0.36651292058166435f;
      tile[M * DSTRIDE + m] = dist * texp - gum;
    }
    // wave-synchronous cross-lane RAW through wave-private LDS region
    asm volatile("s_wait_dscnt 0x0" ::: "memory");
    // lane (m,hi) consumes row m, columns [hi*8, hi*8+8)
#pragma unroll
    for (int u = 0; u < 8; ++u) {
      const float c = tile[m * DSTRIDE + hi * 8 + u];
      const int jj = j0 + hi * 8 + u;
      if (c < bestv[KSEL - 1]) {  // rare after warm-up
        bestv[KSEL - 1] = c;
        bestj[KSEL - 1] = jj;
#pragma unroll
        for (int t = KSEL - 1; t > 0; --t) {
          if (bestv[t] < bestv[t - 1]) {
            const float tv = bestv[t]; bestv[t] = bestv[t - 1]; bestv[t - 1] = tv;
            const int   tj = bestj[t]; bestj[t] = bestj[t - 1]; bestj[t - 1] = tj;
          }
        }
      }
    }
  }

  // Merge the two half-wave lists per row; lane m (hi==0) holds the result
  float ov[KSEL];
  int   oj[KSEL];
#pragma unroll
  for (int t = 0; t < KSEL; ++t) {
    ov[t] = __shfl(bestv[t], m + 16, 32);
    oj[t] = __shfl(bestj[t], m + 16, 32);
  }
  if (hi == 0) {
    float mv[KSEL];
    int   mj[KSEL];
#pragma unroll
    for (int t = 0; t < KSEL; ++t) {
      const bool ta = (bestv[0] <= ov[0]);
      mv[t] = ta ? bestv[0] : ov[0];
      mj[t] = ta ? bestj[0] : oj[0];
      if (ta) {
#pragma unroll
        for (int u2 = 0; u2 < KSEL - 1; ++u2) { bestv[u2] = bestv[u2 + 1]; bestj[u2] = bestj[u2 + 1]; }
        bestv[KSEL - 1] = 3.4e38f;
      } else {
#pragma unroll
        for (int u2 = 0; u2 < KSEL - 1; ++u2) { ov[u2] = ov[u2 + 1]; oj[u2] = oj[u2 + 1]; }
        ov[KSEL - 1] = 3.4e38f;
      }
    }
    const int i = i0 + m;  // global row index (== src edge id)
#pragma unroll
    for (int t = 0; t < KSEL; ++t) {
      lp_out[(size_t)i * KSEL + t]  = -mv[t];
      src_out[(size_t)i * KSEL + t] = (float)i;
      tgt_out[(size_t)i * KSEL + t] = (float)(b * NN + mj[t]);
    }
  }
}

// ---------------------------------------------------------------------------
// Host launcher
// ---------------------------------------------------------------------------
extern "C" void kernel_launch(void* const* d_in, const int* in_sizes, int n_in,
                              void* d_out, int out_size, void* d_ws, size_t ws_size,
                              hipStream_t stream) {
  (void)in_sizes; (void)n_in; (void)out_size; (void)ws_size;
  const float* x    = (const float*)d_in[0];   // [4,4096,64]
  const float* W    = (const float*)d_in[1];   // [64,64]
  const float* temp = (const float*)d_in[2];   // [1]
  const float* q    = (const float*)d_in[3];   // [4,4096,4096]

  float* out  = (float*)d_out;
  float* xemb = out;                               // [b*n*d]
  const size_t XE = (size_t)BB * NN * DD;
  const size_t NE = (size_t)BB * NN * KSEL;
  float* src_out = out + XE;                       // edges row 0
  float* tgt_out = out + XE + NE;                  // edges row 1
  float* lp_out  = out + XE + 2 * NE;              // logprobs

  float* x2 = (float*)d_ws;                        // BB*NN floats

  dgm_embed_wmma<<<(BB * NN) / 16, 128, 0, stream>>>(x, W, xemb);
  dgm_rownorm<<<(BB * NN + 255) / 256, 256, 0, stream>>>(xemb, x2);
  dgm_topk<<<(BB * NN / 16) / 8, 256, 0, stream>>>(xemb, x2, q, temp,
                                                   src_out, tgt_out, lp_out);
}